// GNS_heterogeneous_thesis_49787260895313
// MI455X (gfx1250) — compile-verified
//
#include <hip/hip_runtime.h>
#include <math.h>

typedef float v2f __attribute__((ext_vector_type(2)));
typedef float v8f __attribute__((ext_vector_type(8)));

#define NBUS   20000
#define NGEN   5000
#define NEBB   250000
#define NEGB   5000
#define ECHUNK 62500   // 4 chunks of the bb edge set

__device__ __forceinline__ unsigned ordf(float x) {
  unsigned u = __float_as_uint(x);
  return (u & 0x80000000u) ? ~u : (u | 0x80000000u);
}
__device__ __forceinline__ float unordf(unsigned e) {
  return (e & 0x80000000u) ? __uint_as_float(e & 0x7fffffffu) : __uint_as_float(~e);
}
__device__ __forceinline__ float lrelu(float x) { return x > 0.f ? x : 0.01f * x; }

// ---------------------------------------------------------------------------
// WMMA fp32 GEMM: C[M,N] = A[M,K] @ B[K,N] + bias, optional leaky-relu.
// Requires K % 4 == 0.  One wave computes a 64x16 strip: 4 M-tiles (16x16)
// sharing one B fragment per K-step -> per step: 2 B loads + 4 A b64 loads
// feed 4 x V_WMMA_F32_16X16X4_F32.  Out-of-range rows are clamped (loads stay
// in-bounds, branchless); stores are guarded at the end.
// ---------------------------------------------------------------------------
__global__ __launch_bounds__(256) void gemm_wmma_f32(
    const float* __restrict__ A, const float* __restrict__ B,
    const float* __restrict__ bias, float* __restrict__ C,
    int M, int N, int K, int act)
{
  int wave = threadIdx.x >> 5;
  int lane = threadIdx.x & 31;
  int tiles_n = N >> 4;
  int tiles_m = (M + 63) >> 6;                 // 64-row supertiles
  int tile = blockIdx.x * 8 + wave;
  if (tile >= tiles_m * tiles_n) return;       // wave-uniform exit
  int tm = tile / tiles_n;
  int tn = tile - tm * tiles_n;
  int ml    = lane & 15;
  int kbase = (lane >> 4) << 1;                // 0 or 2
  int col   = tn * 16 + ml;
  int row0  = tm * 64 + ml;

  const float* arow[4];
#pragma unroll
  for (int t = 0; t < 4; ++t) {
    int r = row0 + 16 * t;
    r = r < M ? r : M - 1;                     // clamp: branchless in-bounds load
    arow[t] = A + (size_t)r * K;
  }

  v8f acc[4];
  float bv = bias[col];
#pragma unroll
  for (int t = 0; t < 4; ++t)
#pragma unroll
    for (int r = 0; r < 8; ++r) acc[t][r] = bv;

  for (int k0 = 0; k0 < K; k0 += 4) {
    int ka = k0 + kbase;
    v2f b;
    b[0] = B[(size_t)ka * N + col];
    b[1] = B[(size_t)(ka + 1) * N + col];
#pragma unroll
    for (int t = 0; t < 4; ++t) {
      v2f a = *(const v2f*)(arow[t] + ka);
      acc[t] = __builtin_amdgcn_wmma_f32_16x16x4_f32(false, a, false, b,
                                                     (short)0, acc[t], false, false);
    }
  }

  if (act == 1) {
#pragma unroll
    for (int t = 0; t < 4; ++t)
#pragma unroll
      for (int r = 0; r < 8; ++r) acc[t][r] = lrelu(acc[t][r]);
  }
#pragma unroll
  for (int t = 0; t < 4; ++t) {
#pragma unroll
    for (int r = 0; r < 8; ++r) {
      int m = tm * 64 + 16 * t + ((lane < 16) ? r : r + 8);
      if (m < M) C[(size_t)m * N + col] = acc[t][r];
    }
  }
}

// ---------------------------------------------------------------------------
// LayerNorm over last dim D (32|D), optional trailing leaky-relu + residual.
// One wave per row.
// ---------------------------------------------------------------------------
__global__ __launch_bounds__(256) void ln_kernel(
    const float* __restrict__ X, const float* __restrict__ g,
    const float* __restrict__ b, const float* __restrict__ res,
    float* __restrict__ Y, int N, int D, int do_leaky)
{
  int wave = threadIdx.x >> 5, lane = threadIdx.x & 31;
  int row = blockIdx.x * 8 + wave;
  if (row >= N) return;
  int per = D >> 5;                       // 2 (D=64) or 8 (D=256)
  float v[8];
  float s = 0.f;
  for (int j = 0; j < per; ++j) { v[j] = X[(size_t)row * D + lane + j * 32]; s += v[j]; }
  for (int o = 16; o; o >>= 1) s += __shfl_xor(s, o, 32);
  float mean = s / (float)D;
  float var = 0.f;
  for (int j = 0; j < per; ++j) { float d = v[j] - mean; var += d * d; }
  for (int o = 16; o; o >>= 1) var += __shfl_xor(var, o, 32);
  float inv = rsqrtf(var / (float)D + 1e-5f);
  for (int j = 0; j < per; ++j) {
    int c = lane + j * 32;
    float y = (v[j] - mean) * inv * g[c] + b[c];
    if (do_leaky) y = lrelu(y);
    if (res) y += res[(size_t)row * D + c];
    Y[(size_t)row * D + c] = y;
  }
}

// Zero attention scratch for one conv.
__global__ __launch_bounds__(256) void init_conv_kernel(
    float* __restrict__ outacc, unsigned* __restrict__ amax,
    float* __restrict__ den, int n_dst)
{
  int i = blockIdx.x * 256 + threadIdx.x;
  if (i < n_dst * 256) outacc[i] = 0.f;
  if (i < n_dst * 4) { amax[i] = 0u; den[i] = 0.f; }
}

// Per-edge attention logits + segment-max (ordered-uint atomicMax).
// One wave per edge; lane L handles channels [8L,8L+8), head = L>>3.
__global__ __launch_bounds__(256) void alpha_kernel(
    const float* __restrict__ Q, const float* __restrict__ Km,
    const int* __restrict__ src, const int* __restrict__ dst,
    const float* __restrict__ Ek, float* __restrict__ alpha,
    unsigned* __restrict__ amax, int e0, int nE)
{
  int wave = threadIdx.x >> 5, lane = threadIdx.x & 31;
  int e = e0 + blockIdx.x * 8 + wave;
  if (e >= e0 + nE) return;
  int s = src[e], d = dst[e];
  int c0 = lane * 8;
  const float4* qp = (const float4*)(Q  + (size_t)d * 256 + c0);
  const float4* kp = (const float4*)(Km + (size_t)s * 256 + c0);
  float4 q0 = qp[0], q1 = qp[1];
  float4 k0 = kp[0], k1 = kp[1];
  float sum;
  if (Ek) {
    const float4* ep = (const float4*)(Ek + (size_t)(e - e0) * 256 + c0);
    float4 e0v = ep[0], e1v = ep[1];
    sum = q0.x * (k0.x + e0v.x) + q0.y * (k0.y + e0v.y)
        + q0.z * (k0.z + e0v.z) + q0.w * (k0.w + e0v.w)
        + q1.x * (k1.x + e1v.x) + q1.y * (k1.y + e1v.y)
        + q1.z * (k1.z + e1v.z) + q1.w * (k1.w + e1v.w);
  } else {
    sum = q0.x * k0.x + q0.y * k0.y + q0.z * k0.z + q0.w * k0.w
        + q1.x * k1.x + q1.y * k1.y + q1.z * k1.z + q1.w * k1.w;
  }
  sum += __shfl_xor(sum, 1, 32);
  sum += __shfl_xor(sum, 2, 32);
  sum += __shfl_xor(sum, 4, 32);
  if ((lane & 7) == 0) {
    int h = lane >> 3;
    float a = sum * 0.125f;                 // / sqrt(64)
    alpha[(size_t)e * 4 + h] = a;
    atomicMax(&amax[d * 4 + h], ordf(a));
  }
}

// ex = exp(alpha - amax[dst]); segment-sum of ex.  Thread per (edge,head).
__global__ __launch_bounds__(256) void exp_kernel(
    float* __restrict__ alpha, const unsigned* __restrict__ amax,
    float* __restrict__ den, const int* __restrict__ dst, int nE)
{
  int i = blockIdx.x * 256 + threadIdx.x;
  if (i >= nE * 4) return;
  int e = i >> 2, h = i & 3;
  int d = dst[e];
  float ex = expf(alpha[i] - unordf(amax[d * 4 + h]));
  alpha[i] = ex;
  atomicAdd(&den[d * 4 + h], ex);
}

// Weighted scatter-add of V (+Ek) into out accumulator.  One wave per edge.
__global__ __launch_bounds__(256) void scatter_kernel(
    const float* __restrict__ V, const int* __restrict__ src,
    const int* __restrict__ dst, const float* __restrict__ Ek,
    const float* __restrict__ ex, const float* __restrict__ den,
    float* __restrict__ outacc, int e0, int nE)
{
  int wave = threadIdx.x >> 5, lane = threadIdx.x & 31;
  int e = e0 + blockIdx.x * 8 + wave;
  if (e >= e0 + nE) return;
  int s = src[e], d = dst[e];
  int h = lane >> 3;
  float w = ex[(size_t)e * 4 + h] / (den[d * 4 + h] + 1e-16f);
  int c0 = lane * 8;
  const float4* vp = (const float4*)(V + (size_t)s * 256 + c0);
  float4 v0 = vp[0], v1 = vp[1];
  float val[8];
  if (Ek) {
    const float4* ep = (const float4*)(Ek + (size_t)(e - e0) * 256 + c0);
    float4 e0v = ep[0], e1v = ep[1];
    val[0] = w * (v0.x + e0v.x); val[1] = w * (v0.y + e0v.y);
    val[2] = w * (v0.z + e0v.z); val[3] = w * (v0.w + e0v.w);
    val[4] = w * (v1.x + e1v.x); val[5] = w * (v1.y + e1v.y);
    val[6] = w * (v1.z + e1v.z); val[7] = w * (v1.w + e1v.w);
  } else {
    val[0] = w * v0.x; val[1] = w * v0.y; val[2] = w * v0.z; val[3] = w * v0.w;
    val[4] = w * v1.x; val[5] = w * v1.y; val[6] = w * v1.z; val[7] = w * v1.w;
  }
  float* op = outacc + (size_t)d * 256 + c0;
#pragma unroll
  for (int j = 0; j < 8; ++j) atomicAdd(op + j, val[j]);
}

// beta-gated skip: res = beta*skip + (1-beta)*out.  One wave per node.
__global__ __launch_bounds__(256) void beta_kernel(
    const float* __restrict__ outacc, const float* __restrict__ S,
    const float* __restrict__ Wb, float* __restrict__ H, int n, int accum)
{
  int wave = threadIdx.x >> 5, lane = threadIdx.x & 31;
  int row = blockIdx.x * 8 + wave;
  if (row >= n) return;
  int c0 = lane * 8;
  const float4* op4 = (const float4*)(outacc + (size_t)row * 256 + c0);
  const float4* sp4 = (const float4*)(S + (size_t)row * 256 + c0);
  const float4* w0p = (const float4*)(Wb + c0);
  const float4* w1p = (const float4*)(Wb + 256 + c0);
  const float4* w2p = (const float4*)(Wb + 512 + c0);
  float o[8], sk[8], wb0[8], wb1[8], wb2[8];
  float4 t;
  t = op4[0]; o[0]=t.x; o[1]=t.y; o[2]=t.z; o[3]=t.w;
  t = op4[1]; o[4]=t.x; o[5]=t.y; o[6]=t.z; o[7]=t.w;
  t = sp4[0]; sk[0]=t.x; sk[1]=t.y; sk[2]=t.z; sk[3]=t.w;
  t = sp4[1]; sk[4]=t.x; sk[5]=t.y; sk[6]=t.z; sk[7]=t.w;
  t = w0p[0]; wb0[0]=t.x; wb0[1]=t.y; wb0[2]=t.z; wb0[3]=t.w;
  t = w0p[1]; wb0[4]=t.x; wb0[5]=t.y; wb0[6]=t.z; wb0[7]=t.w;
  t = w1p[0]; wb1[0]=t.x; wb1[1]=t.y; wb1[2]=t.z; wb1[3]=t.w;
  t = w1p[1]; wb1[4]=t.x; wb1[5]=t.y; wb1[6]=t.z; wb1[7]=t.w;
  t = w2p[0]; wb2[0]=t.x; wb2[1]=t.y; wb2[2]=t.z; wb2[3]=t.w;
  t = w2p[1]; wb2[4]=t.x; wb2[5]=t.y; wb2[6]=t.z; wb2[7]=t.w;
  float dot = 0.f;
#pragma unroll
  for (int j = 0; j < 8; ++j)
    dot += o[j] * wb0[j] + sk[j] * wb1[j] + (o[j] - sk[j]) * wb2[j];
  for (int off = 16; off; off >>= 1) dot += __shfl_xor(dot, off, 32);
  float beta = 1.f / (1.f + expf(-dot));
#pragma unroll
  for (int j = 0; j < 8; ++j) {
    float r = beta * sk[j] + (1.f - beta) * o[j];
    size_t idx = (size_t)row * 256 + c0 + j;
    if (accum) H[idx] += r; else H[idx] = r;
  }
}

// Tiny dense layer (per-thread dot): used for K=2 edge proj and head outputs.
__global__ __launch_bounds__(256) void small_gemm_kernel(
    const float* __restrict__ A, const float* __restrict__ B,
    const float* __restrict__ bias, float* __restrict__ C,
    int M, int N, int K, int act)
{
  int i = blockIdx.x * 256 + threadIdx.x;
  if (i >= M * N) return;
  int m = i / N, nn = i - m * N;
  float s = bias[nn];
  for (int k = 0; k < K; ++k) s += A[(size_t)m * K + k] * B[(size_t)k * N + nn];
  if (act == 1) s = lrelu(s);
  C[i] = s;
}

__global__ __launch_bounds__(256) void busout_init_kernel(
    const float* __restrict__ bt, float* __restrict__ out, int n)
{
  int i = blockIdx.x * 256 + threadIdx.x;
  if (i >= n) return;
  out[4 * i]     = bt[4 * i];
  out[4 * i + 1] = bt[4 * i + 1];
  out[4 * i + 2] = 0.f;
  out[4 * i + 3] = 0.f;
}

// AC power-flow physics, scatter-summed by SOURCE bus.
__global__ __launch_bounds__(256) void phys_kernel(
    const float* __restrict__ bt, const float* __restrict__ ea,
    const int* __restrict__ src, const int* __restrict__ dst,
    float* __restrict__ out, int nE)
{
  int e = blockIdx.x * 256 + threadIdx.x;
  if (e >= nE) return;
  int s = src[e], d = dst[e];
  float G = ea[2 * e], B = ea[2 * e + 1];
  float Vms = bt[4 * s], Vas = bt[4 * s + 1];
  float Vmd = bt[4 * d], Vad = bt[4 * d + 1];
  float th = Vas - Vad, vv = Vms * Vmd;
  float st, ct;
  sincosf(th, &st, &ct);
  float P =  G * Vms * Vms - vv * (G * ct + B * st);
  float Q = -B * Vms * Vms - vv * (G * st - B * ct);
  atomicAdd(&out[4 * s + 2], P);
  atomicAdd(&out[4 * s + 3], Q);
}

// ===========================================================================
extern "C" void kernel_launch(void* const* d_in, const int* in_sizes, int n_in,
                              void* d_out, int out_size, void* d_ws, size_t ws_size,
                              hipStream_t stream)
{
  (void)in_sizes; (void)n_in; (void)out_size;
  const float* x_bus = (const float*)d_in[0];
  const float* x_gen = (const float*)d_in[1];
  const int*   ei_bb = (const int*)d_in[2];
  const float* ea_bb = (const float*)d_in[3];
  const int*   ei_gb = (const int*)d_in[4];
  const int*   ei_bg = (const int*)d_in[5];
  const int *s_bb = ei_bb,        *d_bb = ei_bb + NEBB;
  const int *s_gb = ei_gb,        *d_gb = ei_gb + NEGB;
  const int *s_bg = ei_bg,        *d_bg = ei_bg + NEGB;
  auto P = [&](int i) { return (const float*)d_in[i]; };

  // ---- workspace bump allocator (floats) ----
  float* ws = (float*)d_ws;
  size_t off = 0;
  auto alloc = [&](size_t n) { float* p = ws + off; off += n; return p; };
  float* HB0   = alloc((size_t)NBUS * 256);
  float* HB1   = alloc((size_t)NBUS * 256);
  float* HG0   = alloc((size_t)NGEN * 256);
  float* HG1   = alloc((size_t)NGEN * 256);
  float* EBBF  = alloc((size_t)NEBB * 64);     // projected edge features (64)
  float* QB    = alloc((size_t)NBUS * 256);
  float* KB    = alloc((size_t)NBUS * 256);
  float* VB    = alloc((size_t)NBUS * 256);
  float* SB    = alloc((size_t)NBUS * 256);
  float* OACC  = alloc((size_t)NBUS * 256);
  float* HACC  = alloc((size_t)NBUS * 256);
  float* GACC  = alloc((size_t)NGEN * 256);
  float* ALPHA = alloc((size_t)NEBB * 4);
  unsigned* AMAX = (unsigned*)alloc((size_t)NBUS * 4);
  float* DEN   = alloc((size_t)NBUS * 4);
  float* TMPB  = alloc((size_t)NEBB * 64);     // shared: proj temp / Ek chunk (62500*256)
  float* BUST  = alloc((size_t)NBUS * 4);
  if (ws_size < off * sizeof(float)) return;   // workspace too small: bail safely

  auto gemm = [&](const float* A, const float* B, const float* bias, float* C,
                  int M, int N, int K, int act) {
    // K must be a multiple of 4
    int tiles = ((M + 63) / 64) * (N / 16);
    gemm_wmma_f32<<<(tiles + 7) / 8, 256, 0, stream>>>(A, B, bias, C, M, N, K, act);
  };
  auto ln = [&](const float* X, const float* g, const float* b, const float* res,
                float* Y, int N, int D, int lk) {
    ln_kernel<<<(N + 7) / 8, 256, 0, stream>>>(X, g, b, res, Y, N, D, lk);
  };

  // ---- TransformerConv ----
  auto conv = [&](const float* xs, const float* xd, int ns, int nd, int din,
                  const float* Wq, const float* bq, const float* Wk, const float* bk,
                  const float* Wv, const float* bv, const float* Ws, const float* bs,
                  const float* Wb, const int* src, const int* dst, int nE,
                  const float* efeat, const float* We, const float* be,
                  float* Hout, int accum) {
    gemm(xd, Wq, bq, QB, nd, 256, din, 0);
    gemm(xs, Wk, bk, KB, ns, 256, din, 0);
    gemm(xs, Wv, bv, VB, ns, 256, din, 0);
    gemm(xd, Ws, bs, SB, nd, 256, din, 0);
    init_conv_kernel<<<(nd * 256 + 255) / 256, 256, 0, stream>>>(OACC, AMAX, DEN, nd);
    if (efeat) {
      for (int e0 = 0; e0 < nE; e0 += ECHUNK) {
        int cn = nE - e0 < ECHUNK ? nE - e0 : ECHUNK;
        gemm(efeat + (size_t)e0 * 64, We, be, TMPB, cn, 256, 64, 0);
        alpha_kernel<<<(cn + 7) / 8, 256, 0, stream>>>(QB, KB, src, dst, TMPB, ALPHA, AMAX, e0, cn);
      }
    } else {
      alpha_kernel<<<(nE + 7) / 8, 256, 0, stream>>>(QB, KB, src, dst, nullptr, ALPHA, AMAX, 0, nE);
    }
    exp_kernel<<<(nE * 4 + 255) / 256, 256, 0, stream>>>(ALPHA, AMAX, DEN, dst, nE);
    if (efeat) {
      for (int e0 = 0; e0 < nE; e0 += ECHUNK) {
        int cn = nE - e0 < ECHUNK ? nE - e0 : ECHUNK;
        gemm(efeat + (size_t)e0 * 64, We, be, TMPB, cn, 256, 64, 0);
        scatter_kernel<<<(cn + 7) / 8, 256, 0, stream>>>(VB, src, dst, TMPB, ALPHA, DEN, OACC, e0, cn);
      }
    } else {
      scatter_kernel<<<(nE + 7) / 8, 256, 0, stream>>>(VB, src, dst, nullptr, ALPHA, DEN, OACC, 0, nE);
    }
    beta_kernel<<<(nd + 7) / 8, 256, 0, stream>>>(OACC, SB, Wb, Hout, nd, accum);
  };

  // ---- input projections (proj_bus 86.., proj_edge 92.., proj_gen 98..) ----
  gemm(x_bus, P(90), P(86), TMPB, NBUS, 64, 16, 1);
  gemm(TMPB, P(91), P(87), HB0, NBUS, 64, 64, 0);
  ln(HB0, P(89), P(88), nullptr, HB0, NBUS, 64, 0);

  gemm(x_gen, P(102), P(98), TMPB, NGEN, 64, 8, 1);
  gemm(TMPB, P(103), P(99), HG0, NGEN, 64, 64, 0);
  ln(HG0, P(101), P(100), nullptr, HG0, NGEN, 64, 0);

  // K=2: per-thread kernel (WMMA needs K%4==0; 2 MACs/output anyway)
  small_gemm_kernel<<<((NEBB * 64) + 255) / 256, 256, 0, stream>>>(
      ea_bb, P(96), P(92), TMPB, NEBB, 64, 2, 1);
  gemm(TMPB, P(97), P(93), EBBF, NEBB, 64, 64, 0);
  ln(EBBF, P(95), P(94), nullptr, EBBF, NEBB, 64, 0);

  // ---- layers ----
  float* xb = HB0; float* xbn = HB1;
  float* xg = HG0; float* xgn = HG1;
  for (int lp = 0; lp < 2; ++lp) {
    int din = lp ? 256 : 64;
    int L = 8 + lp * 33;
    // bb: Wb=L+2 We=L+3 Wk=L+4 Wq=L+5 Ws=L+6 Wv=L+7 be=L+8 bk=L+9 bq=L+10 bs=L+11 bv=L+12
    // bg: Wb=L+13 Wk=L+14 Wq=L+15 Ws=L+16 Wv=L+17 bk=L+18 bq=L+19 bs=L+20 bv=L+21
    // g_bus=L+22 g_gen=L+23
    // gb: Wb=L+24 Wk=L+25 Wq=L+26 Ws=L+27 Wv=L+28 bk=L+29 bq=L+30 bs=L+31 bv=L+32
    conv(xb, xb, NBUS, NBUS, din,
         P(L+5), P(L+10), P(L+4), P(L+9), P(L+7), P(L+12), P(L+6), P(L+11),
         P(L+2), s_bb, d_bb, NEBB, EBBF, P(L+3), P(L+8), HACC, 0);
    conv(xg, xb, NGEN, NBUS, din,
         P(L+26), P(L+30), P(L+25), P(L+29), P(L+28), P(L+32), P(L+27), P(L+31),
         P(L+24), s_gb, d_gb, NEGB, nullptr, nullptr, nullptr, HACC, 1);
    conv(xb, xg, NBUS, NGEN, din,
         P(L+15), P(L+19), P(L+14), P(L+18), P(L+17), P(L+21), P(L+16), P(L+20),
         P(L+13), s_bg, d_bg, NEGB, nullptr, nullptr, nullptr, GACC, 0);
    ln(HACC, P(L+22), P(L),   lp ? xb : nullptr, xbn, NBUS, 256, 1);
    ln(GACC, P(L+23), P(L+1), lp ? xg : nullptr, xgn, NGEN, 256, 1);
    float* t;
    t = xb; xb = xbn; xbn = t;
    t = xg; xg = xgn; xgn = t;
  }

  // ---- heads (mlp_bus 74..79, mlp_gen 80..85: b1,b2,bt,g,w1,w2) ----
  float* outf = (float*)d_out;
  gemm(xb, P(78), P(74), TMPB, NBUS, 64, 256, 0);
  ln(TMPB, P(77), P(76), nullptr, TMPB, NBUS, 64, 1);
  small_gemm_kernel<<<(NBUS * 4 + 255) / 256, 256, 0, stream>>>(
      TMPB, P(79), P(75), BUST, NBUS, 4, 64, 0);

  gemm(xg, P(84), P(80), TMPB, NGEN, 64, 256, 0);
  ln(TMPB, P(83), P(82), nullptr, TMPB, NGEN, 64, 1);
  small_gemm_kernel<<<(NGEN + 255) / 256, 256, 0, stream>>>(
      TMPB, P(85), P(81), outf + NBUS * 4, NGEN, 1, 64, 0);

  // ---- power-flow physics into output columns 2,3 ----
  busout_init_kernel<<<(NBUS + 255) / 256, 256, 0, stream>>>(BUST, outf, NBUS);
  phys_kernel<<<(NEBB + 255) / 256, 256, 0, stream>>>(BUST, ea_bb, s_bb, d_bb, outf, NEBB);
}